// RoIBBox_9148280340844
// MI455X (gfx1250) — compile-verified
//
#include <hip/hip_runtime.h>

#define BATCH        8
#define N_ANCHORS    200000
#define PRE_TOPN     6000
#define POST_TOPN    1500
#define SORT_N       8192          // next pow2 >= PRE_TOPN
#define TJ           256           // j-tile for NMS mask
#define NTILE        ((PRE_TOPN + TJ - 1) / TJ)   // 24
#define NW           (NTILE * (TJ / 32))          // 192 mask words per row
#define IOU_THR      0.7f

// ---------------------------------------------------------------------------
// Kernel A: per-batch exact top-6000 by prob (3-level radix select on the
// monotone uint view of non-negative floats), descending bitonic sort in LDS,
// then decode only the selected boxes. One 1024-thread block per batch.
// Cut-bin search is two-level (parallel 32-bin group sums, then <=96 serial
// steps) instead of a 2048-step serial suffix scan.
// ---------------------------------------------------------------------------
__global__ __launch_bounds__(1024) void topk_decode_kernel(
    const float* __restrict__ deltas,   // (B, N, 4)
    const float* __restrict__ probs,    // (B, N)
    const float* __restrict__ anchors,  // (N, 4)
    float4* __restrict__ sel_boxes)     // (B, PRE_TOPN)
{
    __shared__ unsigned skey[SORT_N];
    __shared__ int      sval[SORT_N];
    __shared__ unsigned hist[2048];
    __shared__ unsigned gsum[64];
    __shared__ unsigned s_pref;
    __shared__ int      s_remaining, s_nsel, s_neq;

    const int b   = blockIdx.x;
    const int tid = threadIdx.x;
    const unsigned* keys = (const unsigned*)(probs + (size_t)b * N_ANCHORS);

    if (tid == 0) { s_pref = 0u; s_remaining = PRE_TOPN; }
    __syncthreads();

    // ---- exact k-th key via 11/11/10-bit radix refinement ----
    int prefBits = 0;
    const int widths[3] = {11, 11, 10};
    for (int lvl = 0; lvl < 3; ++lvl) {
        const int width = widths[lvl];
        const int nb = 1 << width;
        for (int i = tid; i < nb; i += 1024) hist[i] = 0u;
        __syncthreads();
        const unsigned pref = s_pref;
        for (int t = tid; t < N_ANCHORS; t += 1024) {
            __builtin_prefetch(&keys[t + 8192], 0, 0);   // global_prefetch_b8
            const unsigned k = keys[t];
            if (prefBits == 0 || (k >> (32 - prefBits)) == pref) {
                const unsigned bin = (k >> (32 - prefBits - width)) & (nb - 1);
                atomicAdd(&hist[bin], 1u);
            }
        }
        __syncthreads();

        // parallel 32-bin group sums
        const int ngroups = nb >> 5;
        if (tid < ngroups) {
            unsigned s = 0;
            #pragma unroll 8
            for (int k = 0; k < 32; ++k) s += hist[(tid << 5) + k];
            gsum[tid] = s;
        }
        __syncthreads();

        if (tid == 0) {
            const int remaining = s_remaining;
            // find the group containing the cut (scanning from the top)
            unsigned cum = 0; int g = ngroups - 1;
            for (; g > 0; --g) {
                if (cum + gsum[g] >= (unsigned)remaining) break;
                cum += gsum[g];
            }
            // find the cut bin within group g
            int cut = g << 5; unsigned above = cum;
            for (int bin = (g << 5) + 31; bin >= (g << 5); --bin) {
                const unsigned c = hist[bin];
                if (above + c >= (unsigned)remaining) { cut = bin; break; }
                above += c;
            }
            s_remaining = remaining - (int)above;
            s_pref = (s_pref << width) | (unsigned)cut;
        }
        __syncthreads();
        prefBits += width;
    }

    if (tid == 0) { s_nsel = 0; s_neq = 0; }
    __syncthreads();
    const unsigned T = s_pref;          // exact 6000th-largest key
    const int takeEq = s_remaining;     // #ties at T to keep

    // ---- gather the exact top-6000 (ties resolved by arrival order) ----
    for (int t = tid; t < N_ANCHORS; t += 1024) {
        const unsigned k = keys[t];
        if (k > T) {
            const int p = atomicAdd(&s_nsel, 1);
            if (p < SORT_N) { skey[p] = k; sval[p] = t; }
        } else if (k == T) {
            const int q = atomicAdd(&s_neq, 1);
            if (q < takeEq) {
                const int p = atomicAdd(&s_nsel, 1);
                if (p < SORT_N) { skey[p] = k; sval[p] = t; }
            }
        }
    }
    __syncthreads();
    int nsel = s_nsel; if (nsel > PRE_TOPN) nsel = PRE_TOPN;
    for (int t = nsel + tid; t < SORT_N; t += 1024) { skey[t] = 0u; sval[t] = -1; }
    __syncthreads();

    // ---- descending bitonic sort of (key, idx) in LDS ----
    for (int k = 2; k <= SORT_N; k <<= 1) {
        for (int j = k >> 1; j > 0; j >>= 1) {
            for (int t = tid; t < SORT_N; t += 1024) {
                const int ixj = t ^ j;
                if (ixj > t) {
                    const unsigned a = skey[t], c = skey[ixj];
                    const bool desc = ((t & k) == 0);
                    const bool sw = desc ? (a < c) : (a > c);
                    if (sw) {
                        skey[t] = c; skey[ixj] = a;
                        const int va = sval[t]; sval[t] = sval[ixj]; sval[ixj] = va;
                    }
                }
            }
            __syncthreads();
        }
    }

    // ---- decode only the 6000 selected boxes ----
    for (int r = tid; r < PRE_TOPN; r += 1024) {
        const int a = sval[r];
        float4 o = make_float4(0.f, 0.f, 0.f, 0.f);
        if (a >= 0) {
            const float4 dl = ((const float4*)deltas)[(size_t)b * N_ANCHORS + a];
            const float4 an = ((const float4*)anchors)[a];
            const float aw  = an.w - an.y;
            const float ah  = an.z - an.x;
            const float acx = an.y + 0.5f * aw;
            const float acy = an.x + 0.5f * ah;
            const float dy = dl.x * 0.1f, dx = dl.y * 0.1f;
            const float dh = dl.z * 0.2f, dw = dl.w * 0.2f;
            const float bw = expf(dw) * aw;
            const float bh = expf(dh) * ah;
            const float bcx = dx * aw + acx;
            const float bcy = dy * ah + acy;
            const float y1 = bcy - 0.5f * bh;
            const float x1 = bcx - 0.5f * bw;
            o = make_float4(y1, x1, bh + y1, bw + x1);
        }
        sel_boxes[(size_t)b * PRE_TOPN + r] = o;
    }
}

// ---------------------------------------------------------------------------
// Kernel B: suppression bitmask. Thread = row i; j-tiles of 256 boxes staged
// to LDS via gfx1250 async global->LDS B128 loads (ASYNCcnt path).
// mask[b][i][w] bit l set iff j = 32*w + l suppresses-candidate of i
// (j > i and IoU(i,j) > 0.7). Division-free IoU test.
// ---------------------------------------------------------------------------
__global__ __launch_bounds__(256) void nms_mask_kernel(
    const float4* __restrict__ sel_boxes, unsigned* __restrict__ mask)
{
    __shared__ float4 jb[TJ];
    const int b   = blockIdx.y;
    const int tid = threadIdx.x;
    const int i   = blockIdx.x * 256 + tid;

    float4 bi = make_float4(0.f, 0.f, 0.f, 0.f);
    float areai = 0.f;
    if (i < PRE_TOPN) {
        bi = sel_boxes[(size_t)b * PRE_TOPN + i];
        areai = (bi.z - bi.x) * (bi.w - bi.y);
    }

    for (int tile = 0; tile < NTILE; ++tile) {
        const int j0 = tile * TJ;
        __syncthreads();                       // LDS reuse guard
        const int j = j0 + tid;
        if (j < PRE_TOPN) {
            const unsigned lds_off = (unsigned)(size_t)(&jb[tid]);
            const unsigned long long gaddr =
                (unsigned long long)(const void*)(&sel_boxes[(size_t)b * PRE_TOPN + j]);
            asm volatile("global_load_async_to_lds_b128 %0, %1, off"
                         :: "v"(lds_off), "v"(gaddr) : "memory");
        }
        asm volatile("s_wait_asynccnt 0x0" ::: "memory");
        __syncthreads();

        if (i < PRE_TOPN) {
            #pragma unroll
            for (int w = 0; w < TJ / 32; ++w) {
                unsigned bits = 0u;
                #pragma unroll 8
                for (int l = 0; l < 32; ++l) {
                    const int jj = j0 + w * 32 + l;
                    const float4 bj = jb[w * 32 + l];
                    const float iy1 = fmaxf(bi.x, bj.x);
                    const float ix1 = fmaxf(bi.y, bj.y);
                    const float iy2 = fminf(bi.z, bj.z);
                    const float ix2 = fminf(bi.w, bj.w);
                    const float inter = fmaxf(iy2 - iy1, 0.f) * fmaxf(ix2 - ix1, 0.f);
                    const float areaj = (bj.z - bj.x) * (bj.w - bj.y);
                    const bool sup = (jj > i) && (jj < PRE_TOPN) &&
                                     (inter > IOU_THR * (areai + areaj - inter));
                    bits |= (sup ? 1u : 0u) << l;
                }
                mask[((size_t)b * PRE_TOPN + i) * NW + (size_t)(j0 >> 5) + w] = bits;
            }
        }
    }
}

// ---------------------------------------------------------------------------
// Kernel C: greedy suppression scan — ONE wave32 per batch, wave-synchronous
// (single-wave workgroup: barrier ops degrade to S_NOP; __syncthreads() kept
// purely as a compiler memory fence for the LDS bitmap). Boxes processed in
// 32-wide chunks matching the wave width: only alive boxes are visited, and
// intra-chunk suppression is resolved from the mask row's own chunk word.
// Kept boxes are clipped and written to the output as they are found.
// ---------------------------------------------------------------------------
__global__ __launch_bounds__(32) void nms_scan_kernel(
    const float4* __restrict__ sel_boxes, const unsigned* __restrict__ mask,
    float4* __restrict__ out)
{
    __shared__ unsigned remv[NW];
    const int b    = blockIdx.x;
    const int lane = threadIdx.x;

    for (int w = lane; w < NW; w += 32) remv[w] = 0u;

    int nkeep = 0;
    const int nchunks = (PRE_TOPN + 31) / 32;   // 188
    for (int c = 0; c < nchunks; ++c) {
        __syncthreads();                         // compiler fence (S_NOP in HW)
        unsigned alive = ~remv[c];
        const int base = c * 32;
        if (base + 32 > PRE_TOPN) alive &= (1u << (PRE_TOPN - base)) - 1u;

        while (alive) {
            const int l = __ffs(alive) - 1;      // lowest surviving box
            const int i = base + l;
            const unsigned* row = &mask[((size_t)b * PRE_TOPN + i) * NW];

            if (lane == 0) {                     // emit kept box, clipped
                const float4 bx = sel_boxes[(size_t)b * PRE_TOPN + i];
                float4 o;
                o.x = fminf(fmaxf(bx.x, 0.f), 1.f);
                o.y = fminf(fmaxf(bx.y, 0.f), 1.f);
                o.z = fminf(fmaxf(bx.z, 0.f), 1.f);
                o.w = fminf(fmaxf(bx.w, 0.f), 1.f);
                out[(size_t)b * POST_TOPN + nkeep] = o;
            }
            nkeep++;

            // resolve remaining boxes of this chunk directly
            alive &= ~row[c];
            alive &= ~(1u << l);

            // OR the row into the bitmap for future chunks (6 words/lane)
            for (int w = c + 1 + lane; w < NW; w += 32) remv[w] |= row[w];

            if (nkeep >= POST_TOPN) break;
        }
        if (nkeep >= POST_TOPN) break;
    }

    // zero padding for ranks [nkeep, 1500)
    for (int r = nkeep + lane; r < POST_TOPN; r += 32) {
        out[(size_t)b * POST_TOPN + r] = make_float4(0.f, 0.f, 0.f, 0.f);
    }
}

// ---------------------------------------------------------------------------
extern "C" void kernel_launch(void* const* d_in, const int* in_sizes, int n_in,
                              void* d_out, int out_size, void* d_ws, size_t ws_size,
                              hipStream_t stream) {
    const float* deltas  = (const float*)d_in[0];   // (8, 200000, 4)
    const float* probs   = (const float*)d_in[1];   // (8, 200000)
    const float* anchors = (const float*)d_in[2];   // (200000, 4)
    float4* out          = (float4*)d_out;          // (8, 1500, 4)

    // workspace layout: sel_boxes (768 KB) | mask (36.9 MB)
    float4*   sel_boxes = (float4*)d_ws;
    unsigned* mask      = (unsigned*)((char*)d_ws +
                            (size_t)BATCH * PRE_TOPN * sizeof(float4));

    topk_decode_kernel<<<BATCH, 1024, 0, stream>>>(deltas, probs, anchors, sel_boxes);

    dim3 mgrid((PRE_TOPN + 255) / 256, BATCH);
    nms_mask_kernel<<<mgrid, 256, 0, stream>>>(sel_boxes, mask);

    nms_scan_kernel<<<BATCH, 32, 0, stream>>>(sel_boxes, mask, out);
}